// CompGCNBase_28054726377497
// MI455X (gfx1250) — compile-verified
//
#include <hip/hip_runtime.h>
#include <hip/hip_bf16.h>
#include <cstdint>
#include <cstddef>

// ---------------------------------------------------------------------------
// CompGCN 2-layer forward for MI455X (gfx1250, wave32).
// Algebraic rewrite: scatter-add composed messages (memory-bound, ~1.6 GB/layer
// at 23.3 TB/s HBM), then dense GEMM via V_WMMA_F32_16X16X4_F32 with 5-way
// M-register-blocking for weight-fragment reuse.
// DIM padded 200 -> 208 (13 x 16 tiles); N = 100000 = 80 x 1250 exactly.
// ---------------------------------------------------------------------------

typedef float v2f __attribute__((ext_vector_type(2)));
typedef float v8f __attribute__((ext_vector_type(8)));

#define N_ENT    100000
#define N_RELT   100        // 2 * NUM_REL
#define E_TOT    1000000
#define E_HALF_C 500000
#define DIMC     200
#define LDP      208        // padded leading dim = 13*16
#define NT_TILES 13
#define MB       5          // M-tiles per wave in the fused GEMM (M = 80)
#define MBLKS    1250       // 100000 / 80
#define REL_ROWS 112        // 101 rows padded to 7*16
#define BATCH_C  2048
#define BN_EPS_C 1e-5f

// ------------------------------- utility kernels ---------------------------

__global__ void k_zero4(float4* __restrict__ p, size_t n4) {
  size_t i = (size_t)blockIdx.x * blockDim.x + threadIdx.x;
  if (i < n4) p[i] = make_float4(0.f, 0.f, 0.f, 0.f);
}

__global__ void k_pad_w(const float* __restrict__ w, float* __restrict__ wp) {
  int i = blockIdx.x * blockDim.x + threadIdx.x;
  if (i >= LDP * LDP) return;
  int r = i / LDP, c = i % LDP;
  wp[i] = (r < DIMC && c < DIMC) ? w[r * DIMC + c] : 0.0f;
}

__global__ void k_pad_v(const float* __restrict__ v, float* __restrict__ vp) {
  int c = blockIdx.x * blockDim.x + threadIdx.x;
  if (c < LDP) vp[c] = (c < DIMC) ? v[c] : 0.0f;
}

// x0 = concat(id_embed, gender_w[f0], age_w[f1], level_w[f2]) padded to LDP
__global__ void k_build_x0(const float* __restrict__ id_e, const float* __restrict__ gw,
                           const float* __restrict__ aw, const float* __restrict__ lw,
                           const int* __restrict__ feat, float* __restrict__ x) {
  size_t i = (size_t)blockIdx.x * blockDim.x + threadIdx.x;
  if (i >= (size_t)N_ENT * LDP) return;
  int r = (int)(i / LDP), c = (int)(i % LDP);
  float v = 0.0f;
  if      (c < 50)  v = id_e[(size_t)r * 50 + c];
  else if (c < 100) v = gw[feat[r * 3 + 0] * 50 + (c - 50)];
  else if (c < 150) v = aw[feat[r * 3 + 1] * 50 + (c - 100)];
  else if (c < 200) v = lw[feat[r * 3 + 2] * 50 + (c - 150)];
  x[i] = v;
}

// rel_all = [rel(100 rows) ; loop_rel ; zero-pad] in [REL_ROWS, LDP]
__global__ void k_build_rel_all(const float* __restrict__ src, int src_ld,
                                const float* __restrict__ lrPad, float* __restrict__ dst) {
  int i = blockIdx.x * blockDim.x + threadIdx.x;
  if (i >= REL_ROWS * LDP) return;
  int r = i / LDP, c = i % LDP;
  float v = 0.0f;
  if (r < N_RELT)       v = (c < DIMC) ? src[(size_t)r * src_ld + c] : 0.0f;
  else if (r == N_RELT) v = lrPad[c];
  dst[i] = v;
}

// ------------------------------ graph prep ---------------------------------

__global__ void k_degree(const int* __restrict__ ei, float* __restrict__ degIn,
                         float* __restrict__ degOut) {
  int e = blockIdx.x * blockDim.x + threadIdx.x;
  if (e >= E_TOT) return;
  int row = ei[e];
  atomicAdd((e < E_HALF_C ? degIn : degOut) + row, 1.0f);
}

__global__ void k_norm(const int* __restrict__ ei, const float* __restrict__ degIn,
                       const float* __restrict__ degOut, float* __restrict__ nrm) {
  int e = blockIdx.x * blockDim.x + threadIdx.x;
  if (e >= E_TOT) return;
  int row = ei[e], col = ei[E_TOT + e];
  const float* dg = (e < E_HALF_C) ? degIn : degOut;
  float dr = dg[row], dc = dg[col];
  float ir = dr > 0.0f ? rsqrtf(dr) : 0.0f;
  float ic = dc > 0.0f ? rsqrtf(dc) : 0.0f;
  nrm[e] = ir * ic;
}

// acc[row] += norm[e] * x[col] * rel_all[etype].
// Wave-per-edge: edge metadata loaded once per wave (wave-uniform -> L0
// broadcast), payload as float4 (b128 loads), 4 atomic f32 adds per chunk.
#define EDGES_PER_BLK 8
__global__ __launch_bounds__(256)
void k_compose_scatter(const int* __restrict__ ei, const int* __restrict__ et,
                       const float* __restrict__ nrm, const float* __restrict__ x,
                       const float* __restrict__ relAll,
                       float* __restrict__ accIn, float* __restrict__ accOut) {
  int e = blockIdx.x * EDGES_PER_BLK + (threadIdx.x >> 5);
  if (e >= E_TOT) return;
  int lane = threadIdx.x & 31;
  int row = ei[e], col = ei[E_TOT + e], t = et[e];
  float nv = nrm[e];
  float* acc = (e < E_HALF_C) ? accIn : accOut;
  const float4* xr = (const float4*)(x + (size_t)col * LDP);
  const float4* rr = (const float4*)(relAll + (size_t)t * LDP);
  float* ar = acc + (size_t)row * LDP;
  for (int cc = lane; cc < DIMC / 4; cc += 32) {  // 50 float4 chunks
    float4 xv = xr[cc];
    float4 rv = rr[cc];
    atomicAdd(&ar[cc * 4 + 0], xv.x * rv.x * nv);
    atomicAdd(&ar[cc * 4 + 1], xv.y * rv.y * nv);
    atomicAdd(&ar[cc * 4 + 2], xv.z * rv.z * nv);
    atomicAdd(&ar[cc * 4 + 3], xv.w * rv.w * nv);
  }
}

// ------------------------------ WMMA GEMMs ---------------------------------
// V_WMMA_F32_16X16X4_F32 fragments per CDNA5 ISA 7.12.2:
//  A 16x4 : lane l holds A[l%16, 2*(l/16)+{0,1}]
//  B 4x16 : lane l holds B[2*(l/16)+{0,1}, l%16]
//  C 16x16: VGPR v holds row v (lanes 0-15) / v+8 (lanes 16-31), col = l%16
// Fused kernel: one wave computes an 80x16 output slab (5 M-tiles); the three
// B fragments per K-step are loaded once and reused by 15 WMMAs.

__global__ __launch_bounds__(32)
void k_wmma_fused3(const float* __restrict__ accIn, const float* __restrict__ accOut,
                   const float* __restrict__ x, const float* __restrict__ lrPad,
                   const float* __restrict__ wIn, const float* __restrict__ wOut,
                   const float* __restrict__ wLoop, const float* __restrict__ biasPad,
                   float* __restrict__ pre) {
  int tile = blockIdx.x;
  int nt = tile % NT_TILES, mtb = tile / NT_TILES;
  int lane = threadIdx.x;
  int ncol = nt * 16 + (lane & 15);
  int khalf = (lane >> 4) * 2;
  int mlane = lane & 15;
  int mbase0 = mtb * (16 * MB);
  v8f c[MB];
  #pragma unroll
  for (int mb = 0; mb < MB; ++mb) c[mb] = (v8f){0.f,0.f,0.f,0.f,0.f,0.f,0.f,0.f};

  for (int k0 = 0; k0 < LDP; k0 += 4) {
    int ka = k0 + khalf;
    size_t rb0 = (size_t)ka * LDP + ncol, rb1 = rb0 + LDP;
    v2f bIn  = {wIn[rb0],   wIn[rb1]};
    v2f bOut = {wOut[rb0],  wOut[rb1]};
    v2f bLp  = {wLoop[rb0], wLoop[rb1]};
    float lr0 = lrPad[ka], lr1 = lrPad[ka + 1];
    #pragma unroll
    for (int mb = 0; mb < MB; ++mb) {
      size_t ra = (size_t)(mbase0 + mb * 16 + mlane) * LDP + ka;
      v2f aIn  = {accIn[ra],  accIn[ra + 1]};
      v2f aOut = {accOut[ra], accOut[ra + 1]};
      v2f aLp  = {x[ra] * lr0, x[ra + 1] * lr1};
      c[mb] = __builtin_amdgcn_wmma_f32_16x16x4_f32(false, aIn,  false, bIn,  (short)0, c[mb], false, false);
      c[mb] = __builtin_amdgcn_wmma_f32_16x16x4_f32(false, aOut, false, bOut, (short)0, c[mb], false, false);
      c[mb] = __builtin_amdgcn_wmma_f32_16x16x4_f32(false, aLp,  false, bLp,  (short)0, c[mb], false, false);
    }
  }
  float bcol = biasPad[ncol];
  #pragma unroll
  for (int mb = 0; mb < MB; ++mb) {
    int mrow = mbase0 + mb * 16 + ((lane >> 4) << 3);
    #pragma unroll
    for (int v = 0; v < 8; ++v) {
      pre[(size_t)(mrow + v) * LDP + ncol] = c[mb][v] * (1.0f / 3.0f) + bcol;
    }
  }
}

// plain C = A @ W  for the relation transform, A in [REL_ROWS, LDP]
__global__ __launch_bounds__(32)
void k_wmma_gemm(const float* __restrict__ A, const float* __restrict__ W,
                 float* __restrict__ C) {
  int tile = blockIdx.x;
  int nt = tile % NT_TILES, mt = tile / NT_TILES;
  int lane = threadIdx.x;
  int mrow = mt * 16 + (lane & 15);
  int ncol = nt * 16 + (lane & 15);
  int khalf = (lane >> 4) * 2;
  v8f c = {0.f, 0.f, 0.f, 0.f, 0.f, 0.f, 0.f, 0.f};
  for (int k0 = 0; k0 < LDP; k0 += 4) {
    int ka = k0 + khalf;
    size_t ra = (size_t)mrow * LDP + ka;
    v2f a = {A[ra], A[ra + 1]};
    v2f b = {W[(size_t)ka * LDP + ncol], W[(size_t)(ka + 1) * LDP + ncol]};
    c = __builtin_amdgcn_wmma_f32_16x16x4_f32(false, a, false, b, (short)0, c, false, false);
  }
  int mbase = mt * 16 + ((lane >> 4) << 3);
  #pragma unroll
  for (int v = 0; v < 8; ++v) {
    C[(size_t)(mbase + v) * LDP + ncol] = c[v];
  }
}

// ------------------------------ batch norm ---------------------------------
// One thread per column (blockDim = LDP): consecutive threads read consecutive
// addresses -> fully coalesced row-major sweep; one atomicAdd per block/col.

#define BN_ROWS 500
__global__ __launch_bounds__(LDP)
void k_bn_reduce(const float* __restrict__ pre, float* __restrict__ colSum,
                 float* __restrict__ colSq) {
  int c = threadIdx.x;
  int r0 = blockIdx.x * BN_ROWS;
  int r1 = r0 + BN_ROWS; if (r1 > N_ENT) r1 = N_ENT;
  float a = 0.0f, b = 0.0f;
  for (int r = r0; r < r1; ++r) {
    float v = pre[(size_t)r * LDP + c];
    a += v; b += v * v;
  }
  atomicAdd(&colSum[c], a);
  atomicAdd(&colSq[c], b);
}

__global__ void k_bn_final(const float* __restrict__ colSum, const float* __restrict__ colSq,
                           float* __restrict__ meanB, float* __restrict__ invB) {
  int c = blockIdx.x * blockDim.x + threadIdx.x;
  if (c >= LDP) return;
  float m = colSum[c] / (float)N_ENT;
  float var = colSq[c] / (float)N_ENT - m * m;
  meanB[c] = m;
  invB[c] = rsqrtf(var + BN_EPS_C);
}

__global__ void k_bn_tanh(float* __restrict__ x, const float* __restrict__ meanB,
                          const float* __restrict__ invB, const float* __restrict__ g,
                          const float* __restrict__ bt) {
  size_t i = (size_t)blockIdx.x * blockDim.x + threadIdx.x;
  if (i >= (size_t)N_ENT * LDP) return;
  int c = (int)(i % LDP);
  x[i] = tanhf(g[c] * (x[i] - meanB[c]) * invB[c] + bt[c]);
}

// ------------------------------ outputs ------------------------------------

__global__ void k_out_gather(const float* __restrict__ src, const int* __restrict__ idx,
                             float* __restrict__ out, int rows) {
  int i = blockIdx.x * blockDim.x + threadIdx.x;
  if (i >= rows * DIMC) return;
  int r = i / DIMC, c = i % DIMC;
  out[i] = src[(size_t)idx[r] * LDP + c];
}

__global__ void k_out_x(const float* __restrict__ x, float* __restrict__ out) {
  size_t i = (size_t)blockIdx.x * blockDim.x + threadIdx.x;
  if (i >= (size_t)N_ENT * DIMC) return;
  size_t r = i / DIMC, c = i % DIMC;
  out[i] = x[r * LDP + c];
}

// ------------------------------ host launcher ------------------------------

static inline unsigned gdiv(size_t n, unsigned b) { return (unsigned)((n + b - 1) / b); }

extern "C" void kernel_launch(void* const* d_in, const int* in_sizes, int n_in,
                              void* d_out, int out_size, void* d_ws, size_t ws_size,
                              hipStream_t stream) {
  (void)in_sizes; (void)n_in; (void)out_size; (void)ws_size;
  const int*   edge_index = (const int*)d_in[0];   // [2, 1e6]
  const int*   edge_type  = (const int*)d_in[1];   // [1e6]
  const int*   ent_feat   = (const int*)d_in[2];   // [1e5, 3]
  const int*   sub        = (const int*)d_in[3];   // [2048]
  const int*   rel        = (const int*)d_in[4];   // [2048]
  const float* id_embed   = (const float*)d_in[5];
  const float* gender_w   = (const float*)d_in[6];
  const float* age_w      = (const float*)d_in[7];
  const float* level_w    = (const float*)d_in[8];
  const float* init_rel   = (const float*)d_in[9];
  // per-layer params: w_in, w_out, w_loop, w_rel, loop_rel, b, gamma, beta
  const float* P[2][8];
  for (int li = 0; li < 2; ++li)
    for (int j = 0; j < 8; ++j) P[li][j] = (const float*)d_in[10 + li * 8 + j];

  float* ws = (float*)d_ws;
  size_t f = 0;
  float* xA     = ws + f; f += (size_t)N_ENT * LDP;
  float* xB     = ws + f; f += (size_t)N_ENT * LDP;
  float* accIn  = ws + f; f += (size_t)N_ENT * LDP;
  float* accOut = ws + f; f += (size_t)N_ENT * LDP;
  float* nrm    = ws + f; f += (size_t)E_TOT;
  float* degIn  = ws + f; f += (size_t)N_ENT;
  float* degOut = ws + f; f += (size_t)N_ENT;
  float* wpad   = ws + f; f += (size_t)8 * LDP * LDP;  // win1,wout1,wloop1,wrel1,win2,...
  float* relAll1  = ws + f; f += (size_t)REL_ROWS * LDP;
  float* relOut1  = ws + f; f += (size_t)REL_ROWS * LDP;
  float* relAll2  = ws + f; f += (size_t)REL_ROWS * LDP;
  float* relFinal = ws + f; f += (size_t)REL_ROWS * LDP;
  float* bpad   = ws + f; f += 2 * LDP;
  float* gpad   = ws + f; f += 2 * LDP;
  float* btpad  = ws + f; f += 2 * LDP;
  float* lrpad  = ws + f; f += 2 * LDP;
  float* colSum = ws + f; f += LDP;
  float* colSq  = ws + f; f += LDP;
  float* meanB  = ws + f; f += LDP;
  float* invB   = ws + f; f += LDP;

  const unsigned TB = 256;
  const size_t NLD = (size_t)N_ENT * LDP;

  // ---- per-call prep: padded weights/vectors, x0, degrees, norms
  for (int li = 0; li < 2; ++li)
    for (int j = 0; j < 4; ++j)
      k_pad_w<<<gdiv(LDP * LDP, TB), TB, 0, stream>>>(P[li][j], wpad + (size_t)(li * 4 + j) * LDP * LDP);
  for (int li = 0; li < 2; ++li) {
    k_pad_v<<<1, LDP, 0, stream>>>(P[li][4], lrpad + li * LDP);  // loop_rel
    k_pad_v<<<1, LDP, 0, stream>>>(P[li][5], bpad  + li * LDP);  // b
    k_pad_v<<<1, LDP, 0, stream>>>(P[li][6], gpad  + li * LDP);  // gamma
    k_pad_v<<<1, LDP, 0, stream>>>(P[li][7], btpad + li * LDP);  // beta
  }
  k_build_x0<<<gdiv(NLD, TB), TB, 0, stream>>>(id_embed, gender_w, age_w, level_w, ent_feat, xA);
  k_build_rel_all<<<gdiv(REL_ROWS * LDP, TB), TB, 0, stream>>>(init_rel, DIMC, lrpad, relAll1);
  k_zero4<<<gdiv((size_t)2 * N_ENT / 4, TB), TB, 0, stream>>>((float4*)degIn, (size_t)2 * N_ENT / 4);
  k_degree<<<gdiv(E_TOT, TB), TB, 0, stream>>>(edge_index, degIn, degOut);
  k_norm<<<gdiv(E_TOT, TB), TB, 0, stream>>>(edge_index, degIn, degOut, nrm);

  float* xcur = xA;
  float* xnxt = xB;
  for (int li = 0; li < 2; ++li) {
    const float* relAll = (li == 0) ? relAll1 : relAll2;
    const float* wIn    = wpad + (size_t)(li * 4 + 0) * LDP * LDP;
    const float* wOut   = wpad + (size_t)(li * 4 + 1) * LDP * LDP;
    const float* wLoop  = wpad + (size_t)(li * 4 + 2) * LDP * LDP;
    const float* wRel   = wpad + (size_t)(li * 4 + 3) * LDP * LDP;

    // scatter composed messages (accIn and accOut are contiguous: zero both)
    k_zero4<<<gdiv(2 * NLD / 4, TB), TB, 0, stream>>>((float4*)accIn, 2 * NLD / 4);
    k_compose_scatter<<<gdiv(E_TOT, EDGES_PER_BLK), 256, 0, stream>>>(
        edge_index, edge_type, nrm, xcur, relAll, accIn, accOut);

    // fused triple GEMM + bias (WMMA, 80x16 slab per wave)
    k_wmma_fused3<<<MBLKS * NT_TILES, 32, 0, stream>>>(
        accIn, accOut, xcur, lrpad + li * LDP, wIn, wOut, wLoop, bpad + li * LDP, xnxt);

    // batch-norm (training stats) + tanh, in place on xnxt
    k_zero4<<<gdiv(2 * LDP / 4, TB), TB, 0, stream>>>((float4*)colSum, 2 * LDP / 4);
    k_bn_reduce<<<gdiv(N_ENT, BN_ROWS), LDP, 0, stream>>>(xnxt, colSum, colSq);
    k_bn_final<<<1, LDP, 0, stream>>>(colSum, colSq, meanB, invB);
    k_bn_tanh<<<gdiv(NLD, TB), TB, 0, stream>>>(xnxt, meanB, invB, gpad + li * LDP, btpad + li * LDP);

    // relation transform: rel_next = rel_all @ w_rel (rows 0..99 used)
    float* rdst = (li == 0) ? relOut1 : relFinal;
    k_wmma_gemm<<<(REL_ROWS / 16) * NT_TILES, 32, 0, stream>>>(relAll, wRel, rdst);
    if (li == 0)
      k_build_rel_all<<<gdiv(REL_ROWS * LDP, TB), TB, 0, stream>>>(relOut1, LDP, lrpad + LDP, relAll2);

    float* t = xcur; xcur = xnxt; xnxt = t;
  }

  // outputs: [sub_emb 2048x200 | rel_emb 2048x200 | x 100000x200]
  float* out = (float*)d_out;
  k_out_gather<<<gdiv((size_t)BATCH_C * DIMC, TB), TB, 0, stream>>>(xcur, sub, out, BATCH_C);
  k_out_gather<<<gdiv((size_t)BATCH_C * DIMC, TB), TB, 0, stream>>>(relFinal, rel,
                                                                    out + (size_t)BATCH_C * DIMC, BATCH_C);
  k_out_x<<<gdiv((size_t)N_ENT * DIMC, TB), TB, 0, stream>>>(xcur, out + 2 * (size_t)BATCH_C * DIMC);
}